// WildcatPool2d_31207232372919
// MI455X (gfx1250) — compile-verified
//
#include <hip/hip_runtime.h>
#include <stdint.h>

// ---------------------------------------------------------------------------
// WildcatPool2d on MI455X (gfx1250, wave32)
//   x : [32, 56, 56, 512] f32   ->  out : [32, 512] f32
//   out[b,c] = 0.5 * ( sum(top627)/627 + 0.7*sum(bottom627)/627 )
// Exact radix-select (11/11/10-bit digits) on monotone uint keys, data held
// in VGPRs, LDS histograms, async LDS->global store of the results.
// ---------------------------------------------------------------------------

#define B_DIM    32
#define C_DIM    512
#define N_SP     3136      // 56*56
#define K_SEL    627       // round(0.2 * 3136)
#define ALPHA_F  0.7f
#define CH_WG    16        // channels per workgroup
#define THREADS  256
#define QV       49        // float4 loads per thread: 3136*16/(4*256)
#define HSTRIDE  2068      // padded per-channel hist stride (words); 2068%64=20
#define NB1      2048      // 11-bit digit
#define NB2      2048      // 11-bit digit
#define NB3      1024      // 10-bit digit

// LDS word layout
#define OFF_HIST   0
#define OFF_PARTS  (CH_WG * HSTRIDE)                 // 16*16 chunk partials
#define OFF_RANKHI (OFF_PARTS + 256)
#define OFF_DIGHI  (OFF_RANKHI + CH_WG)
#define OFF_RANKLO (OFF_DIGHI + CH_WG)
#define OFF_DIGLO  (OFF_RANKLO + CH_WG)
#define OFF_PREFHI (OFF_DIGLO + CH_WG)
#define OFF_PREFLO (OFF_PREFHI + CH_WG)
#define OFF_CNTHI  (OFF_PREFLO + CH_WG)              // uint counts (atomic)
#define OFF_CNTLO  (OFF_CNTHI + CH_WG)
#define OFF_PSUMHI (OFF_CNTLO + CH_WG)               // 16ch x 64 partial sums
#define OFF_PSUMLO (OFF_PSUMHI + CH_WG * 64)
#define OFF_RES    (OFF_PSUMLO + CH_WG * 64)
#define SMEM_WORDS (OFF_RES + CH_WG)
#define SMEM_BYTES (SMEM_WORDS * 4)

__device__ __forceinline__ uint32_t fkey(uint32_t u) {
  // order-preserving float->uint map
  return u ^ (uint32_t)(((int32_t)u >> 31) | 0x80000000);
}
__device__ __forceinline__ float fdecode(uint32_t k) {
  uint32_t u = (k & 0x80000000u) ? (k ^ 0x80000000u) : ~k;
  return __uint_as_float(u);
}

// Cooperative digit scan over per-channel histograms.
// rank[c] is 1-based rank from top (fromTop) or bottom; on exit:
// digit[c] = digit containing the rank-th element, rank[c] = rank within it.
__device__ void scan_digit(uint32_t* hist, uint32_t* parts,
                           uint32_t* rank, uint32_t* digit,
                           int nbins, bool fromTop) {
  const int tid   = threadIdx.x;
  const int ch    = tid >> 4;     // 16 threads per channel
  const int part  = tid & 15;
  const int chunk = nbins >> 4;
  const uint32_t* h = hist + ch * HSTRIDE + part * chunk;
  uint32_t s = 0;
  for (int i = 0; i < chunk; ++i) s += h[i];
  parts[(ch << 4) + part] = s;
  __syncthreads();
  if (tid < CH_WG) {
    const int c = tid;
    uint32_t k = rank[c];
    const uint32_t* pp = parts + (c << 4);
    const uint32_t* hh = hist + c * HSTRIDE;
    uint32_t cum = 0;
    int d;
    if (fromTop) {
      int p = 15;
      while (p > 0 && cum + pp[p] < k) { cum += pp[p]; --p; }
      d = (p + 1) * chunk - 1;
      const int lo = p * chunk;
      while (d > lo && cum + hh[d] < k) { cum += hh[d]; --d; }
    } else {
      int p = 0;
      while (p < 15 && cum + pp[p] < k) { cum += pp[p]; ++p; }
      d = p * chunk;
      const int hi = (p + 1) * chunk - 1;
      while (d < hi && cum + hh[d] < k) { cum += hh[d]; ++d; }
    }
    rank[c]  = k - cum;      // >= 1, rank inside digit d
    digit[c] = (uint32_t)d;
  }
  __syncthreads();
}

extern "C" __global__ void __launch_bounds__(THREADS, 1)
wildcat_pool2d_kernel(const float* __restrict__ x, float* __restrict__ out) {
  extern __shared__ uint32_t smem[];
  uint32_t* hist   = smem + OFF_HIST;
  uint32_t* parts  = smem + OFF_PARTS;
  uint32_t* rankHi = smem + OFF_RANKHI;
  uint32_t* digHi  = smem + OFF_DIGHI;
  uint32_t* rankLo = smem + OFF_RANKLO;
  uint32_t* digLo  = smem + OFF_DIGLO;
  uint32_t* prefHi = smem + OFF_PREFHI;
  uint32_t* prefLo = smem + OFF_PREFLO;
  uint32_t* cntHiA = smem + OFF_CNTHI;
  uint32_t* cntLoA = smem + OFF_CNTLO;
  float*    psumHi = (float*)(smem + OFF_PSUMHI);
  float*    psumLo = (float*)(smem + OFF_PSUMLO);
  float*    resBuf = (float*)(smem + OFF_RES);

  const int tid   = threadIdx.x;
  const int b     = blockIdx.x >> 5;            // 32 channel-blocks per batch
  const int c0    = (blockIdx.x & 31) * CH_WG;
  const int j4    = tid & 3;                    // fixed 4-channel sub-quad
  const int chb   = j4 << 2;                    // local channel base (0,4,8,12)
  const int srow0 = tid >> 2;                   // 0..63

  const float* base = x + (size_t)b * N_SP * C_DIM + c0;

  if (tid < CH_WG) {
    rankHi[tid] = K_SEL; rankLo[tid] = K_SEL;
    prefHi[tid] = 0u;    prefLo[tid] = 0u;
    cntHiA[tid] = 0u;    cntLoA[tid] = 0u;
  }
  for (int idx = tid; idx < CH_WG * NB1; idx += THREADS)
    hist[(idx >> 11) * HSTRIDE + (idx & (NB1 - 1))] = 0u;

  // Warm GL2 for this workgroup's slab (global_prefetch_b8).
  #pragma unroll
  for (int i = 0; i < QV; ++i) {
    int s = (i << 6) + srow0;
    __builtin_prefetch((const void*)((const float4*)(base + (size_t)s * C_DIM) + j4), 0, 0);
  }

  // Single HBM pass: coalesced b128 loads; keep monotone keys in VGPRs.
  uint32_t key[QV * 4];
  #pragma unroll
  for (int i = 0; i < QV; ++i) {
    int s = (i << 6) + srow0;
    float4 v = *((const float4*)(base + (size_t)s * C_DIM) + j4);
    key[i * 4 + 0] = fkey(__float_as_uint(v.x));
    key[i * 4 + 1] = fkey(__float_as_uint(v.y));
    key[i * 4 + 2] = fkey(__float_as_uint(v.z));
    key[i * 4 + 3] = fkey(__float_as_uint(v.w));
  }
  __syncthreads();

  // ---- pass 1: 11-bit digit histogram (shared between hi and lo selects) --
  #pragma unroll
  for (int i = 0; i < QV; ++i) {
    #pragma unroll
    for (int j = 0; j < 4; ++j)
      atomicAdd(&hist[(chb + j) * HSTRIDE + (key[i * 4 + j] >> 21)], 1u);
  }
  __syncthreads();
  scan_digit(hist, parts, rankHi, digHi, NB1, true);
  scan_digit(hist, parts, rankLo, digLo, NB1, false);
  if (tid < CH_WG) { prefHi[tid] = digHi[tid] << 21; prefLo[tid] = digLo[tid] << 21; }
  __syncthreads();

  // ---- hi refine: bits [20:10] --------------------------------------------
  for (int idx = tid; idx < CH_WG * NB2; idx += THREADS)
    hist[(idx >> 11) * HSTRIDE + (idx & (NB2 - 1))] = 0u;
  __syncthreads();
  {
    uint32_t p[4];
    #pragma unroll
    for (int j = 0; j < 4; ++j) p[j] = prefHi[chb + j] >> 21;
    #pragma unroll
    for (int i = 0; i < QV; ++i) {
      #pragma unroll
      for (int j = 0; j < 4; ++j) {
        uint32_t k = key[i * 4 + j];
        if ((k >> 21) == p[j])
          atomicAdd(&hist[(chb + j) * HSTRIDE + ((k >> 10) & 0x7FFu)], 1u);
      }
    }
  }
  __syncthreads();
  scan_digit(hist, parts, rankHi, digHi, NB2, true);
  if (tid < CH_WG) prefHi[tid] |= digHi[tid] << 10;
  __syncthreads();

  // ---- hi refine: bits [9:0] ----------------------------------------------
  for (int idx = tid; idx < CH_WG * NB3; idx += THREADS)
    hist[(idx >> 10) * HSTRIDE + (idx & (NB3 - 1))] = 0u;
  __syncthreads();
  {
    uint32_t p[4];
    #pragma unroll
    for (int j = 0; j < 4; ++j) p[j] = prefHi[chb + j] >> 10;
    #pragma unroll
    for (int i = 0; i < QV; ++i) {
      #pragma unroll
      for (int j = 0; j < 4; ++j) {
        uint32_t k = key[i * 4 + j];
        if ((k >> 10) == p[j])
          atomicAdd(&hist[(chb + j) * HSTRIDE + (k & 0x3FFu)], 1u);
      }
    }
  }
  __syncthreads();
  scan_digit(hist, parts, rankHi, digHi, NB3, true);
  if (tid < CH_WG) prefHi[tid] |= digHi[tid];   // exact key of 627th-largest
  __syncthreads();

  // ---- lo refine: bits [20:10] --------------------------------------------
  for (int idx = tid; idx < CH_WG * NB2; idx += THREADS)
    hist[(idx >> 11) * HSTRIDE + (idx & (NB2 - 1))] = 0u;
  __syncthreads();
  {
    uint32_t p[4];
    #pragma unroll
    for (int j = 0; j < 4; ++j) p[j] = prefLo[chb + j] >> 21;
    #pragma unroll
    for (int i = 0; i < QV; ++i) {
      #pragma unroll
      for (int j = 0; j < 4; ++j) {
        uint32_t k = key[i * 4 + j];
        if ((k >> 21) == p[j])
          atomicAdd(&hist[(chb + j) * HSTRIDE + ((k >> 10) & 0x7FFu)], 1u);
      }
    }
  }
  __syncthreads();
  scan_digit(hist, parts, rankLo, digLo, NB2, false);
  if (tid < CH_WG) prefLo[tid] |= digLo[tid] << 10;
  __syncthreads();

  // ---- lo refine: bits [9:0] ----------------------------------------------
  for (int idx = tid; idx < CH_WG * NB3; idx += THREADS)
    hist[(idx >> 10) * HSTRIDE + (idx & (NB3 - 1))] = 0u;
  __syncthreads();
  {
    uint32_t p[4];
    #pragma unroll
    for (int j = 0; j < 4; ++j) p[j] = prefLo[chb + j] >> 10;
    #pragma unroll
    for (int i = 0; i < QV; ++i) {
      #pragma unroll
      for (int j = 0; j < 4; ++j) {
        uint32_t k = key[i * 4 + j];
        if ((k >> 10) == p[j])
          atomicAdd(&hist[(chb + j) * HSTRIDE + (k & 0x3FFu)], 1u);
      }
    }
  }
  __syncthreads();
  scan_digit(hist, parts, rankLo, digLo, NB3, false);
  if (tid < CH_WG) prefLo[tid] |= digLo[tid];   // exact key of 627th-smallest
  __syncthreads();

  // ---- final predicated sums (registers) ----------------------------------
  uint32_t TH[4], TL[4];
  #pragma unroll
  for (int j = 0; j < 4; ++j) { TH[j] = prefHi[chb + j]; TL[j] = prefLo[chb + j]; }
  float sH[4] = {0, 0, 0, 0}, sL[4] = {0, 0, 0, 0};
  uint32_t cH[4] = {0, 0, 0, 0}, cL[4] = {0, 0, 0, 0};
  #pragma unroll
  for (int i = 0; i < QV; ++i) {
    #pragma unroll
    for (int j = 0; j < 4; ++j) {
      uint32_t k = key[i * 4 + j];
      float f = fdecode(k);
      if (k > TH[j]) { sH[j] += f; cH[j] += 1u; }
      if (k < TL[j]) { sL[j] += f; cL[j] += 1u; }
    }
  }
  // counts: integer atomics (deterministic); sums: fixed-order tree in LDS
  #pragma unroll
  for (int j = 0; j < 4; ++j) {
    atomicAdd(&cntHiA[chb + j], cH[j]);
    atomicAdd(&cntLoA[chb + j], cL[j]);
    psumHi[(chb + j) * 64 + srow0] = sH[j];
    psumLo[(chb + j) * 64 + srow0] = sL[j];
  }
  __syncthreads();
  #pragma unroll
  for (int s = 32; s >= 1; s >>= 1) {
    for (int idx = tid; idx < CH_WG * s; idx += THREADS) {
      int c = idx / s, m = idx % s;          // s is a constant per unrolled step
      psumHi[c * 64 + m] += psumHi[c * 64 + m + s];
      psumLo[c * 64 + m] += psumLo[c * 64 + m + s];
    }
    __syncthreads();
  }

  if (tid < CH_WG) {
    const float kf = (float)K_SEL;
    float vH = fdecode(prefHi[tid]);
    float vL = fdecode(prefLo[tid]);
    float top = (psumHi[tid * 64] + (kf - (float)cntHiA[tid]) * vH) * (1.0f / kf);
    float bot = (psumLo[tid * 64] + (kf - (float)cntLoA[tid]) * vL) * (ALPHA_F / kf);
    resBuf[tid] = 0.5f * (top + bot);
  }
  __syncthreads();

  // ---- CDNA5 async store: LDS -> global, fenced by split counters ---------
  if (tid < CH_WG) {
    asm volatile("s_wait_dscnt 0x0" ::: "memory");
    const float* gaddr = out + (size_t)b * C_DIM + c0 + tid;
    uint32_t ldsoff = (uint32_t)(uintptr_t)(&resBuf[tid]);
    asm volatile("global_store_async_from_lds_b32 %0, %1, off"
                 :: "v"(gaddr), "v"(ldsoff) : "memory");
    asm volatile("s_wait_asynccnt 0x0" ::: "memory");
  }
}

extern "C" void kernel_launch(void* const* d_in, const int* in_sizes, int n_in,
                              void* d_out, int out_size, void* d_ws, size_t ws_size,
                              hipStream_t stream) {
  (void)in_sizes; (void)n_in; (void)d_ws; (void)ws_size; (void)out_size;
  const float* x = (const float*)d_in[0];
  float* out = (float*)d_out;
  dim3 grid(B_DIM * (C_DIM / CH_WG));   // 1024 workgroups
  dim3 block(THREADS);                  // 8 waves (wave32)
  wildcat_pool2d_kernel<<<grid, block, SMEM_BYTES, stream>>>(x, out);
}